// ContrastiveLoss_41154376630896
// MI455X (gfx1250) — compile-verified
//
#include <hip/hip_runtime.h>
#include <hip/hip_bf16.h>
#include <math.h>

#define N_ROWS 8192
#define DIMD   512
#define INV_TAU (1.0f / 0.07f)

typedef __attribute__((ext_vector_type(2))) float v2f;
typedef __attribute__((ext_vector_type(8))) float v8f;

// ---------------- zero the row/col accumulators (2*N contiguous floats) ----
__global__ __launch_bounds__(256) void zero_sums(float* __restrict__ sums) {
    int i = blockIdx.x * 256 + threadIdx.x;
    if (i < 2 * N_ROWS) sums[i] = 0.0f;
}

// ---------------- row-normalize z_i and z_j into workspace ----------------
__global__ __launch_bounds__(128) void normalize_rows(const float* __restrict__ zi,
                                                      const float* __restrict__ zj,
                                                      float* __restrict__ zin,
                                                      float* __restrict__ zjn) {
    const float* src = blockIdx.y ? zj : zi;
    float*       dst = blockIdx.y ? zjn : zin;
    const int row = blockIdx.x;
    const int t   = threadIdx.x;            // 128 threads * float4 = 512 floats
    const float4 v = ((const float4*)(src + (size_t)row * DIMD))[t];
    float ss = v.x * v.x + v.y * v.y + v.z * v.z + v.w * v.w;
#pragma unroll
    for (int m = 1; m < 32; m <<= 1) ss += __shfl_xor(ss, m, 32);
    __shared__ float red[4];
    if ((t & 31) == 0) red[t >> 5] = ss;
    __syncthreads();
    const float tot   = red[0] + red[1] + red[2] + red[3];
    const float scale = 1.0f / fmaxf(sqrtf(tot), 1e-8f);
    float4 o;
    o.x = v.x * scale; o.y = v.y * scale; o.z = v.z * scale; o.w = v.w * scale;
    ((float4*)(dst + (size_t)row * DIMD))[t] = o;
}

// ---------------- fused WMMA GEMM + exp + masked row/col reductions -------
#define BM 128
#define BN 128
#define BK 32
#define LDT 36   // padded LDS stride (floats): 144B, 16B-aligned, conflict-free

__global__ __launch_bounds__(256) void simclr_gemm(const float* __restrict__ A,  // zi_n [N,D]
                                                   const float* __restrict__ B,  // zj_n [N,D]
                                                   float* __restrict__ row_sum,
                                                   float* __restrict__ col_sum,
                                                   float* __restrict__ pos) {
    __shared__ float sA[BM * LDT];
    __shared__ float sB[BN * LDT];

    const int tid    = threadIdx.x;
    const int lane   = tid & 31;
    const int wave   = tid >> 5;       // 8 waves
    const int wm     = wave >> 1;      // 0..3 -> 32-row group
    const int wn     = wave & 1;       // 0..1 -> 64-col group
    const int lhalf  = lane >> 4;      // 0/1
    const int l15    = lane & 15;
    const int rowBase = blockIdx.y * BM;
    const int colBase = blockIdx.x * BN;

    v8f acc[2][4];
#pragma unroll
    for (int i = 0; i < 2; ++i)
#pragma unroll
        for (int j = 0; j < 4; ++j)
            acc[i][j] = (v8f){0.f, 0.f, 0.f, 0.f, 0.f, 0.f, 0.f, 0.f};

    // global->LDS staging: thread t -> row t/2, half t&1 (16 floats = 4x float4)
    const int lr = tid >> 1;
    const int lh = tid & 1;
    const float* gA = A + (size_t)(rowBase + lr) * DIMD + lh * 16;
    const float* gB = B + (size_t)(colBase + lr) * DIMD + lh * 16;
    float* sAp = &sA[lr * LDT + lh * 16];
    float* sBp = &sB[lr * LDT + lh * 16];

    for (int k0 = 0; k0 < DIMD; k0 += BK) {
        __syncthreads();
#pragma unroll
        for (int q = 0; q < 4; ++q) {
            const float4 va = *(const float4*)(gA + k0 + q * 4);
            const float4 vb = *(const float4*)(gB + k0 + q * 4);
            *(float4*)(sAp + q * 4) = va;
            *(float4*)(sBp + q * 4) = vb;
        }
        __syncthreads();

#pragma unroll
        for (int kk = 0; kk < BK; kk += 4) {
            const int kr = kk + 2 * lhalf;   // lanes 0-15: K=kk..kk+1, lanes 16-31: K=kk+2..kk+3
            v2f a[2], b[4];
            a[0] = *(const v2f*)&sA[(wm * 32 +      l15) * LDT + kr];
            a[1] = *(const v2f*)&sA[(wm * 32 + 16 + l15) * LDT + kr];
#pragma unroll
            for (int j = 0; j < 4; ++j)
                b[j] = *(const v2f*)&sB[(wn * 64 + j * 16 + l15) * LDT + kr];
#pragma unroll
            for (int i = 0; i < 2; ++i)
#pragma unroll
                for (int j = 0; j < 4; ++j)
                    acc[i][j] = __builtin_amdgcn_wmma_f32_16x16x4_f32(
                        false, a[i], false, b[j], (short)0, acc[i][j], false, false);
        }
    }

    // ---- epilogue: e = exp(sim - 1/tau); diagonal -> pos, excluded ----
    float rowp[2][8];
    float colp[4];
#pragma unroll
    for (int i = 0; i < 2; ++i)
#pragma unroll
        for (int v = 0; v < 8; ++v) rowp[i][v] = 0.f;
#pragma unroll
    for (int j = 0; j < 4; ++j) colp[j] = 0.f;

#pragma unroll
    for (int i = 0; i < 2; ++i) {
#pragma unroll
        for (int j = 0; j < 4; ++j) {
#pragma unroll
            for (int v = 0; v < 8; ++v) {
                const int mg = rowBase + wm * 32 + i * 16 + v + 8 * lhalf;
                const int ng = colBase + wn * 64 + j * 16 + l15;
                float d = acc[i][j][v];
                if (mg == ng) {
                    pos[mg] = d * INV_TAU;
                    d = -INFINITY;           // exclude diagonal: exp -> 0
                }
                const float ex = __expf((d - 1.0f) * INV_TAU);
                rowp[i][v] += ex;
                colp[j]    += ex;
            }
        }
    }

    // column sums: combine the two M-halves, lanes 0-15 own the 16 columns
#pragma unroll
    for (int j = 0; j < 4; ++j) {
        const float c = colp[j] + __shfl_xor(colp[j], 16, 32);
        if (lane < 16) atomicAdd(&col_sum[colBase + wn * 64 + j * 16 + l15], c);
    }
    // row sums: reduce over the 16 N-lanes within each half
#pragma unroll
    for (int i = 0; i < 2; ++i) {
#pragma unroll
        for (int v = 0; v < 8; ++v) {
            float r = rowp[i][v];
            r += __shfl_xor(r, 1, 32);
            r += __shfl_xor(r, 2, 32);
            r += __shfl_xor(r, 4, 32);
            r += __shfl_xor(r, 8, 32);
            if (l15 == 0)
                atomicAdd(&row_sum[rowBase + wm * 32 + i * 16 + v + 8 * lhalf], r);
        }
    }
}

// ---------------- finalize: lse = 1/tau + log(sum); means; 3 outputs ------
__global__ __launch_bounds__(256) void finalize(const float* __restrict__ row_sum,
                                                const float* __restrict__ col_sum,
                                                const float* __restrict__ pos,
                                                float* __restrict__ out) {
    const int t = threadIdx.x;
    float se = 0.f, st = 0.f;
    for (int i = t; i < N_ROWS; i += 256) {
        const float p = pos[i];
        se += INV_TAU + logf(row_sum[i]) - p;
        st += INV_TAU + logf(col_sum[i]) - p;
    }
#pragma unroll
    for (int m = 1; m < 32; m <<= 1) {
        se += __shfl_xor(se, m, 32);
        st += __shfl_xor(st, m, 32);
    }
    __shared__ float rse[8], rst[8];
    if ((t & 31) == 0) { rse[t >> 5] = se; rst[t >> 5] = st; }
    __syncthreads();
    if (t == 0) {
        float S = 0.f, T = 0.f;
#pragma unroll
        for (int w = 0; w < 8; ++w) { S += rse[w]; T += rst[w]; }
        const float le = S / (float)N_ROWS;
        const float lt = T / (float)N_ROWS;
        out[0] = 0.5f * (le + lt);
        out[1] = le;
        out[2] = lt;
    }
}

extern "C" void kernel_launch(void* const* d_in, const int* in_sizes, int n_in,
                              void* d_out, int out_size, void* d_ws, size_t ws_size,
                              hipStream_t stream) {
    const float* zi = (const float*)d_in[0];
    const float* zj = (const float*)d_in[1];
    float* ws  = (float*)d_ws;
    float* zin = ws;
    float* zjn = zin + (size_t)N_ROWS * DIMD;
    float* row_sum = zjn + (size_t)N_ROWS * DIMD;
    float* col_sum = row_sum + N_ROWS;   // contiguous with row_sum (zeroed together)
    float* pos     = col_sum + N_ROWS;
    float* out = (float*)d_out;

    zero_sums<<<dim3((2 * N_ROWS) / 256), dim3(256), 0, stream>>>(row_sum);
    normalize_rows<<<dim3(N_ROWS, 2), dim3(128), 0, stream>>>(zi, zj, zin, zjn);
    simclr_gemm<<<dim3(N_ROWS / BN, N_ROWS / BM), dim3(256), 0, stream>>>(
        zin, zjn, row_sum, col_sum, pos);
    finalize<<<dim3(1), dim3(256), 0, stream>>>(row_sum, col_sum, pos, out);
}